// InvariantPointAttentionV39_32512902430728
// MI455X (gfx1250) — compile-verified
//
#include <hip/hip_runtime.h>
#include <hip/hip_bf16.h>

// ---------------------------------------------------------------------------
// Invariant Point Attention, fused for MI455X (gfx1250, wave32, WMMA).
// HBM-bound on the 302MB `pair` tensor: stream it once (non-temporal) into a
// [H,N,N] bias buffer (28MB, L2-resident), then fuse logits+softmax+attn@v
// per (head, 16-row tile) with the row block in LDS. All matmuls use
// V_WMMA_F32_16X16X4_F32 (fp32-exact). No predication inside WMMA k-loops;
// GEMMs are templated on compile-time dims so inner loops use immediate
// load offsets (no per-iteration 64-bit address math).
// ---------------------------------------------------------------------------

#define N_    768
#define C_    384
#define PC_   128
#define H_    12
#define CH_   32
#define NN_   (N_ * N_)
#define HP_   108            // H*3*P raw point width
#define HPP_  144            // H*12 padded point width
#define SCALE_ 0.17677669529663687f   // 32^-0.5
#define EPS_   1e-5f

typedef float v2f __attribute__((ext_vector_type(2)));
typedef float v8f __attribute__((ext_vector_type(8)));

// D = A(16x4 f32) * B(4x16 f32) + C  -- wave32 WMMA
__device__ __forceinline__ v8f wmma4(v2f a, v2f b, v8f c) {
  return __builtin_amdgcn_wmma_f32_16x16x4_f32(
      /*neg_a=*/false, a, /*neg_b=*/false, b,
      /*c_mod=*/(short)0, c, /*reuse_a=*/false, /*reuse_b=*/false);
}

// ---------------------------------------------------------------------------
// C[M,NCOLS] = A[M,K] @ B[K,NCOLS] + bias.  One wave per 16x16 tile.
// All dims compile-time -> immediate global-load offsets. Out-of-range
// columns are CLAMPED for loads (they only pollute unstored D columns).
// ---------------------------------------------------------------------------
template <int K, int LDA, int LDB, int LDC, int NCOLS>
__global__ __launch_bounds__(256) void ipa_gemm_bias(
    const float* __restrict__ A, const float* __restrict__ B,
    const float* __restrict__ bias, float* __restrict__ C)
{
  const int lane = threadIdx.x & 31;
  const int wave = threadIdx.x >> 5;
  const int half = lane >> 4;
  const int r    = lane & 15;
  const int row0 = blockIdx.x * 16;
  const int col0 = (blockIdx.y * 8 + wave) * 16;
  if (col0 >= NCOLS) return;                 // wave-uniform exit
  const int  colN  = col0 + r;
  const bool colOK = (colN < NCOLS);
  const int  colL  = colOK ? colN : (NCOLS - 1);   // clamped load column

  const float* Ap = A + (row0 + r) * LDA + 2 * half;
  const float* Bp = B + colL + 2 * half * LDB;
  v8f acc = {};
#pragma unroll 8
  for (int k0 = 0; k0 < K; k0 += 4) {
    v2f a; a.x = Ap[k0];           a.y = Ap[k0 + 1];
    v2f b; b.x = Bp[k0 * LDB];     b.y = Bp[(k0 + 1) * LDB];
    acc = wmma4(a, b, acc);
  }
  const float bv = colOK ? bias[colL] : 0.0f;
#pragma unroll
  for (int i = 0; i < 8; ++i) {
    const int row = row0 + i + 8 * half;
    if (colOK) C[row * LDC + colN] = acc[i] + bv;
  }
}

// ---------------------------------------------------------------------------
// pair_bias[h, n, m] = pair[n, m, :] . Wpb[:, h] + bpb[h]
// A = Wpb^T staged in LDS as [head][k] with 132-float pitch (conflict-free
// ds_load_b64), B = pair rows (302MB NT stream, immediate offsets).
// D rows = heads, D cols = 16 consecutive (n,m) rows -> coalesced stores.
// ---------------------------------------------------------------------------
__global__ __launch_bounds__(256) void ipa_pair_bias(
    const float* __restrict__ pair, const float* __restrict__ Wpb,
    const float* __restrict__ bpb, float* __restrict__ bias)
{
  const int lane = threadIdx.x & 31;
  const int wave = threadIdx.x >> 5;
  const int half = lane >> 4;
  const int r    = lane & 15;

  __shared__ float sW[H_ * 132];   // [head][k], pitch 132 (~6.2KB)
  __shared__ float sB[H_];
  for (int t = threadIdx.x; t < PC_ * H_; t += 256) {
    const int kk = t / H_, hh = t - kk * H_;
    sW[hh * 132 + kk] = Wpb[t];
  }
  if (threadIdx.x < H_) sB[threadIdx.x] = bpb[threadIdx.x];
  __syncthreads();

  const int row0 = (blockIdx.x * 8 + wave) * 16;   // row = n*768 + m
  const int m    = (r < H_) ? r : (H_ - 1);        // clamped head (A row)
  const float* Prow = pair + (row0 + r) * PC_ + 2 * half;
  const float* Wrow = &sW[m * 132 + 2 * half];

  v8f acc = {};
#pragma unroll
  for (int k0 = 0; k0 < PC_; k0 += 4) {
    v2f a = *(const v2f*)(Wrow + k0);                        // ds_load_b64
    v2f b = __builtin_nontemporal_load((const v2f*)(Prow + k0));
    acc = wmma4(a, b, acc);
  }
#pragma unroll
  for (int i = 0; i < 8; ++i) {
    const int h = i + 8 * half;                    // D row = head
    if (h < H_)
      bias[h * NN_ + row0 + r] = acc[i] + sB[h];
  }
}

// ---------------------------------------------------------------------------
// Rigid transform of points + squared norms, writing zero-padded [N,H,12]
// layout so the attention kernel needs no K-guards.
// g[n,h,d,j] = sum_i pts[n,h,d,i]*rot[n,i,j] + trans[n,d];  s2[n,h]=sum g^2
// ---------------------------------------------------------------------------
__global__ void ipa_rot_pts(
    const float* __restrict__ qp, const float* __restrict__ kp,
    const float* __restrict__ rot, const float* __restrict__ trans,
    float* __restrict__ qg, float* __restrict__ kg,
    float* __restrict__ q2, float* __restrict__ k2)
{
  const int idx = blockIdx.x * blockDim.x + threadIdx.x;   // n*H + h
  if (idx >= N_ * H_) return;
  const int n = idx / H_;
  const int h = idx - n * H_;
  float R[3][3], t[3];
#pragma unroll
  for (int i = 0; i < 3; ++i) {
    t[i] = trans[n * 3 + i];
#pragma unroll
    for (int j = 0; j < 3; ++j) R[i][j] = rot[n * 9 + i * 3 + j];
  }
  const float* ins[2]   = {qp, kp};
  float*       gouts[2] = {qg, kg};
  float*       souts[2] = {q2, k2};
#pragma unroll
  for (int s = 0; s < 2; ++s) {
    const float* p = ins[s] + n * HP_ + h * 9;
    float*       g = gouts[s] + n * HPP_ + h * 12;
    float loc[9];
#pragma unroll
    for (int e = 0; e < 9; ++e) loc[e] = p[e];
    float acc2 = 0.0f;
#pragma unroll
    for (int d = 0; d < 3; ++d)
#pragma unroll
      for (int j = 0; j < 3; ++j) {
        float gv = loc[d * 3 + 0] * R[0][j] + loc[d * 3 + 1] * R[1][j] +
                   loc[d * 3 + 2] * R[2][j] + t[d];
        g[d * 3 + j] = gv;
        acc2 += gv * gv;
      }
    g[9] = 0.0f; g[10] = 0.0f; g[11] = 0.0f;
    souts[s][n * H_ + h] = acc2;
  }
}

// ---------------------------------------------------------------------------
// Fused attention per (head, 16-row n-tile):
//   logits[16,768] = SCALE*(q.k^T + qg.kg^T) - 0.5*SCALE*(q2+k2) + bias
//   softmax over m, then weighted[16,32] = attn @ v_h (split-K over 8 waves,
//   reduced with LDS float atomics).
// ---------------------------------------------------------------------------
__global__ __launch_bounds__(256) void ipa_attn(
    const float* __restrict__ q,  const float* __restrict__ k,
    const float* __restrict__ v,  const float* __restrict__ qg,
    const float* __restrict__ kg, const float* __restrict__ q2,
    const float* __restrict__ k2, const float* __restrict__ bias,
    float* __restrict__ weighted)
{
  const int h    = blockIdx.y;
  const int n0   = blockIdx.x * 16;
  const int lane = threadIdx.x & 31;
  const int wave = threadIdx.x >> 5;
  const int half = lane >> 4;
  const int r    = lane & 15;

  __shared__ float sL[16][772];      // logits / probs (772: conflict-free)
  __shared__ float sAcc[16][32];     // attn@v reduction

  // ---- Phase 1: logits, 16x16 tiles over m, 6 tiles per wave ----
  const float* qrow  = q  + (n0 + r) * C_ + h * CH_ + 2 * half;
  const float* qgrow = qg + (n0 + r) * HPP_ + h * 12 + 2 * half;
  float q2v[8];
#pragma unroll
  for (int i = 0; i < 8; ++i) q2v[i] = q2[(n0 + i + 8 * half) * H_ + h];

  for (int mt = wave; mt < N_ / 16; mt += 8) {
    const int mcol = mt * 16 + r;
    v8f acc = {};
    // scalar q.k^T, K = 32
    const float* krow = k + mcol * C_ + h * CH_ + 2 * half;
#pragma unroll
    for (int k0 = 0; k0 < CH_; k0 += 4) {
      v2f a; a.x = qrow[k0];  a.y = qrow[k0 + 1];
      v2f b; b.x = krow[k0];  b.y = krow[k0 + 1];
      acc = wmma4(a, b, acc);
    }
    // point qg.kg^T, K = 12 (zero-padded, unguarded)
    const float* kgrow = kg + mcol * HPP_ + h * 12 + 2 * half;
#pragma unroll
    for (int k0 = 0; k0 < 12; k0 += 4) {
      v2f a; a.x = qgrow[k0]; a.y = qgrow[k0 + 1];
      v2f b; b.x = kgrow[k0]; b.y = kgrow[k0 + 1];
      acc = wmma4(a, b, acc);
    }
    const float k2m = k2[mcol * H_ + h];
    const float* brow = bias + h * NN_ + mcol;
#pragma unroll
    for (int i = 0; i < 8; ++i) {
      const int nrow = i + 8 * half;
      sL[nrow][mcol] = SCALE_ * acc[i]
                     - 0.5f * SCALE_ * (q2v[i] + k2m)
                     + brow[(n0 + nrow) * N_];
    }
  }
  __syncthreads();

  // ---- Phase 2: softmax, 2 rows per wave, lane-parallel over 768 ----
#pragma unroll
  for (int rr = 0; rr < 2; ++rr) {
    const int row = wave * 2 + rr;
    float mx = -1e30f;
    for (int c = lane; c < N_; c += 32) mx = fmaxf(mx, sL[row][c]);
#pragma unroll
    for (int off = 16; off; off >>= 1) mx = fmaxf(mx, __shfl_xor(mx, off, 32));
    float sum = 0.0f;
    for (int c = lane; c < N_; c += 32) {
      float e = __expf(sL[row][c] - mx);
      sL[row][c] = e;
      sum += e;
    }
#pragma unroll
    for (int off = 16; off; off >>= 1) sum += __shfl_xor(sum, off, 32);
    const float inv = 1.0f / sum;
    for (int c = lane; c < N_; c += 32) sL[row][c] *= inv;
  }
  __syncthreads();
  for (int t = threadIdx.x; t < 16 * 32; t += 256) (&sAcc[0][0])[t] = 0.0f;
  __syncthreads();

  // ---- Phase 3: attn @ v, split-K: wave owns m in [wave*96, wave*96+96) ----
  v8f acc0 = {}, acc1 = {};
  const int kbase = wave * 96;
  const float* aL = &sL[r][kbase + 2 * half];
  const float* vB = v + (kbase + 2 * half) * C_ + h * CH_;
#pragma unroll 4
  for (int kk = 0; kk < 96; kk += 4) {
    v2f a; a.x = aL[kk]; a.y = aL[kk + 1];
    const float* vr0 = vB + kk * C_;
    v2f b0; b0.x = vr0[r];      b0.y = vr0[C_ + r];
    acc0 = wmma4(a, b0, acc0);
    v2f b1; b1.x = vr0[16 + r]; b1.y = vr0[C_ + 16 + r];
    acc1 = wmma4(a, b1, acc1);
  }
#pragma unroll
  for (int i = 0; i < 8; ++i) {
    const int row = i + 8 * half;
    atomicAdd(&sAcc[row][r],      acc0[i]);
    atomicAdd(&sAcc[row][16 + r], acc1[i]);
  }
  __syncthreads();
  for (int t = threadIdx.x; t < 512; t += 256) {
    const int row = t >> 5, col = t & 31;
    weighted[(n0 + row) * C_ + h * CH_ + col] = sAcc[row][col];
  }
}

// ---------------------------------------------------------------------------
// out = LN(single + weighted @ Wo + bo) * gamma + beta
// ---------------------------------------------------------------------------
__global__ __launch_bounds__(256) void ipa_out_ln(
    const float* __restrict__ weighted, const float* __restrict__ Wo,
    const float* __restrict__ bo, const float* __restrict__ single,
    const float* __restrict__ gamma, const float* __restrict__ beta,
    float* __restrict__ out)
{
  const int n0   = blockIdx.x * 16;
  const int lane = threadIdx.x & 31;
  const int wave = threadIdx.x >> 5;
  const int half = lane >> 4;
  const int r    = lane & 15;

  __shared__ float sX[16][388];

  const float* Ap = weighted + (n0 + r) * C_ + 2 * half;
#pragma unroll
  for (int ct = 0; ct < 3; ++ct) {
    const int colN = (wave * 3 + ct) * 16 + r;
    const float* Bp = Wo + colN + 2 * half * C_;
    v8f acc = {};
#pragma unroll 8
    for (int k0 = 0; k0 < C_; k0 += 4) {
      v2f a; a.x = Ap[k0];       a.y = Ap[k0 + 1];
      v2f b; b.x = Bp[k0 * C_];  b.y = Bp[(k0 + 1) * C_];
      acc = wmma4(a, b, acc);
    }
#pragma unroll
    for (int i = 0; i < 8; ++i) {
      const int row = i + 8 * half;
      sX[row][colN] = acc[i] + bo[colN] + single[(n0 + row) * C_ + colN];
    }
  }
  __syncthreads();

#pragma unroll
  for (int rr = 0; rr < 2; ++rr) {
    const int row = wave * 2 + rr;
    const int gn  = n0 + row;
    float s = 0.0f;
    for (int c = lane; c < C_; c += 32) s += sX[row][c];
#pragma unroll
    for (int off = 16; off; off >>= 1) s += __shfl_xor(s, off, 32);
    const float mu = s * (1.0f / C_);
    float vs = 0.0f;
    for (int c = lane; c < C_; c += 32) {
      const float d = sX[row][c] - mu;
      vs += d * d;
    }
#pragma unroll
    for (int off = 16; off; off >>= 1) vs += __shfl_xor(vs, off, 32);
    const float inv = rsqrtf(vs * (1.0f / C_) + EPS_);
    for (int c = lane; c < C_; c += 32)
      out[gn * C_ + c] = (sX[row][c] - mu) * inv * gamma[c] + beta[c];
  }
}

// ---------------------------------------------------------------------------
extern "C" void kernel_launch(void* const* d_in, const int* in_sizes, int n_in,
                              void* d_out, int out_size, void* d_ws, size_t ws_size,
                              hipStream_t stream) {
  const float* single = (const float*)d_in[0];
  const float* pair   = (const float*)d_in[1];
  const float* rot    = (const float*)d_in[2];
  const float* trans  = (const float*)d_in[3];
  const float* Wq  = (const float*)d_in[4];
  const float* bq  = (const float*)d_in[5];
  const float* Wk  = (const float*)d_in[6];
  const float* bk  = (const float*)d_in[7];
  const float* Wv  = (const float*)d_in[8];
  const float* bv  = (const float*)d_in[9];
  const float* Wpb = (const float*)d_in[10];
  const float* bpb = (const float*)d_in[11];
  const float* Wqp = (const float*)d_in[12];
  const float* bqp = (const float*)d_in[13];
  const float* Wkp = (const float*)d_in[14];
  const float* bkp = (const float*)d_in[15];
  const float* Wo  = (const float*)d_in[16];
  const float* bo  = (const float*)d_in[17];
  const float* gamma = (const float*)d_in[18];
  const float* beta  = (const float*)d_in[19];
  float* out = (float*)d_out;

  // workspace layout (floats)
  float* wq  = (float*)d_ws;                 // 768*384
  float* wk  = wq  + N_ * C_;
  float* wv  = wk  + N_ * C_;
  float* wqp = wv  + N_ * C_;                // raw pts 768*108
  float* wkp = wqp + N_ * HP_;
  float* wqg = wkp + N_ * HP_;               // padded pts 768*144
  float* wkg = wqg + N_ * HPP_;
  float* wq2 = wkg + N_ * HPP_;              // 768*12
  float* wk2 = wq2 + N_ * H_;
  float* wbias = wk2 + N_ * H_;              // 12*768*768
  float* wweighted = wbias + (size_t)H_ * NN_;

  const dim3 blk(256);

  // q / k / v projections: [768,384] @ [384,384]
  ipa_gemm_bias<C_, C_, C_, C_, C_><<<dim3(N_ / 16, 3), blk, 0, stream>>>(single, Wq, bq, wq);
  ipa_gemm_bias<C_, C_, C_, C_, C_><<<dim3(N_ / 16, 3), blk, 0, stream>>>(single, Wk, bk, wk);
  ipa_gemm_bias<C_, C_, C_, C_, C_><<<dim3(N_ / 16, 3), blk, 0, stream>>>(single, Wv, bv, wv);
  // point projections: [768,384] @ [384,108]
  ipa_gemm_bias<C_, C_, HP_, HP_, HP_><<<dim3(N_ / 16, 1), blk, 0, stream>>>(single, Wqp, bqp, wqp);
  ipa_gemm_bias<C_, C_, HP_, HP_, HP_><<<dim3(N_ / 16, 1), blk, 0, stream>>>(single, Wkp, bkp, wkp);
  // pair bias: [NN,128] @ [128,12] -> [12,NN]  (the 302MB NT stream)
  ipa_pair_bias<<<dim3(NN_ / (16 * 8)), blk, 0, stream>>>(pair, Wpb, bpb, wbias);
  // rigid transform + squared norms (writes zero-padded K=12 layout)
  ipa_rot_pts<<<dim3((N_ * H_ + 255) / 256), blk, 0, stream>>>(
      wqp, wkp, rot, trans, wqg, wkg, wq2, wk2);
  // fused attention per (n-tile, head)
  ipa_attn<<<dim3(N_ / 16, H_), blk, 0, stream>>>(wq, wk, wv, wqg, wkg, wq2, wk2, wbias, wweighted);
  // output projection + residual + layernorm
  ipa_out_ln<<<dim3(N_ / 16), blk, 0, stream>>>(wweighted, Wo, bo, single, gamma, beta, out);

  (void)in_sizes; (void)n_in; (void)out_size; (void)ws_size;
}